// RNNDecoder_83270825934911
// MI455X (gfx1250) — compile-verified
//
#include <hip/hip_runtime.h>
#include <math.h>

typedef __attribute__((ext_vector_type(2))) float v2f;
typedef __attribute__((ext_vector_type(8))) float v8f;
typedef __attribute__((ext_vector_type(4))) unsigned int u32x4;
typedef __attribute__((ext_vector_type(8))) int i32x8;
typedef __attribute__((ext_vector_type(4))) int i32x4;

#define HID   1024
#define GATES 4096
#define TLEN  4096
#define VOCAB 50257

#if defined(__has_builtin)
#if __has_builtin(__builtin_amdgcn_tensor_load_to_lds) && __has_builtin(__builtin_amdgcn_s_wait_tensorcnt)
#define USE_TDM 1
#endif
#endif

// ---------------------------------------------------------------------------
// WMMA fp32 GEMV: out[r] = (acc ? out[r] : 0) + (bias ? bias[r] : 0) + W[r,:] @ v
//
// One wave -> 16 output rows. Per 16-column chunk the k-mapping is permuted so
// each half-wave owns 8 consecutive columns (two b128 loads from W and from
// LDS), feeding 4 chained V_WMMA_F32_16X16X4_F32 ops. A and B use the same
// per-lane column mapping, so every column pairs with its vector element
// exactly once; B is replicated across all 16 N-columns so every column of D
// equals the dot product (extraction depends only on the documented D layout).
// The vector is staged into LDS by the Tensor Data Mover when available.
// ---------------------------------------------------------------------------
__global__ __launch_bounds__(256) void gemv_wmma(const float* __restrict__ W,
                                                 const float* __restrict__ v,
                                                 const float* __restrict__ bias,
                                                 float* __restrict__ out,
                                                 int M, int K, int ld, int accumulate)
{
    __shared__ __align__(16) float vlds[2048];
    const int tid = threadIdx.x;

#ifdef USE_TDM
    if (tid < 32) {  // one wave issues the block's DMA; TDM ignores EXEC, per-wave issue
        const unsigned long long ga = (unsigned long long)(uintptr_t)v;
        const unsigned int lds_off = (unsigned int)(uintptr_t)vlds;  // flat addr low 32b = LDS offset
        u32x4 g0;
        g0[0] = 1u;                                                   // count=1, user descriptor
        g0[1] = lds_off;                                              // lds_addr
        g0[2] = (unsigned int)(ga & 0xFFFFFFFFu);                     // global_addr[31:0]
        g0[3] = (unsigned int)((ga >> 32) & 0x1FFFFFFu) | (2u << 30); // global_addr[56:32] | type=2
        i32x8 g1;
        g1[0] = (2 << 16);                               // data_size=4B; no multicast/iterate/pad
        g1[1] = (int)((unsigned)K << 16);                // tensor_dim0[15:0]  @ bits[63:48]
        g1[2] = (int)(((unsigned)K >> 16) | (1u << 16)); // tensor_dim0[31:16] | tensor_dim1=1
        g1[3] = (int)((unsigned)K << 16);                // tile_dim0=K        @ bits[127:112]
        g1[4] = 1;                                       // tile_dim1=1, tile_dim2=0
        g1[5] = K;                                       // tensor_dim0_stride=K (low 32)
        g1[6] = 0;
        g1[7] = 0;
        i32x4 z4 = {0, 0, 0, 0};
#if defined(__clang_major__) && __clang_major__ >= 23
        i32x8 z8 = {};
        __builtin_amdgcn_tensor_load_to_lds(g0, g1, z4, z4, z8, 0);
#else
        __builtin_amdgcn_tensor_load_to_lds(g0, g1, z4, z4, 0);
#endif
        __builtin_amdgcn_s_wait_tensorcnt(0);
    }
#else
    for (int i = tid; i < K; i += 256) vlds[i] = v[i];
#endif
    __syncthreads();

    const int wave = tid >> 5;          // 0..7
    const int lane = tid & 31;
    const int half = lane >> 4;         // half-wave id
    const int mloc = lane & 15;

    const int rowBase = (blockIdx.x * 8 + wave) * 16;
    if (rowBase >= M) return;           // wave-uniform: EXEC stays all-ones for WMMA

    const int row  = rowBase + mloc;
    const int rowc = row < M ? row : (M - 1);   // clamp OOB lanes (dup loads, no store)
    const float* __restrict__ wrow = W + (size_t)rowc * (size_t)ld;
    const int koff = 8 * half;                  // this half-wave owns cols k0+koff .. +7

    v8f acc = {};
    for (int k0 = 0; k0 < K; k0 += 16) {
        if (k0 + 256 < K) __builtin_prefetch(wrow + k0 + 256, 0, 1);  // global_prefetch_b8

        const float4 a0 = *(const float4*)(wrow + k0 + koff);
        const float4 a1 = *(const float4*)(wrow + k0 + koff + 4);
        const float4 b0 = *(const float4*)(vlds + k0 + koff);
        const float4 b1 = *(const float4*)(vlds + k0 + koff + 4);

        v2f a, b;
        a.x = a0.x; a.y = a0.y; b.x = b0.x; b.y = b0.y;
        acc = __builtin_amdgcn_wmma_f32_16x16x4_f32(false, a, false, b, (short)0, acc, false, false);
        a.x = a0.z; a.y = a0.w; b.x = b0.z; b.y = b0.w;
        acc = __builtin_amdgcn_wmma_f32_16x16x4_f32(false, a, false, b, (short)0, acc, false, false);
        a.x = a1.x; a.y = a1.y; b.x = b1.x; b.y = b1.y;
        acc = __builtin_amdgcn_wmma_f32_16x16x4_f32(false, a, false, b, (short)0, acc, false, false);
        a.x = a1.z; a.y = a1.w; b.x = b1.z; b.y = b1.w;
        acc = __builtin_amdgcn_wmma_f32_16x16x4_f32(false, a, false, b, (short)0, acc, false, false);
    }

    // D layout: lanes 0-15 hold rows rowBase+0..7 in acc[0..7]; lanes 16-31 hold +8..15.
    if (lane == 0 || lane == 16) {
        const int base = rowBase + 8 * half;
#pragma unroll
        for (int j = 0; j < 8; ++j) {
            const int r = base + j;
            if (r < M) {
                float val = acc[j];
                if (bias)       val += bias[r];
                if (accumulate) val += out[r];
                out[r] = val;
            }
        }
    }
}

// ---------------------------------------------------------------------------
// LSTM pointwise: gates -> h_new, c_new (fanned out to ws + d_out slots)
// ---------------------------------------------------------------------------
__global__ __launch_bounds__(256) void lstm_pointwise(const float* __restrict__ gates,
                                                      const float* __restrict__ c_old,
                                                      float* __restrict__ h_new,
                                                      float* __restrict__ c_new,
                                                      float* __restrict__ concat_h,
                                                      float* __restrict__ out_h,
                                                      float* __restrict__ out_c)
{
    const int i = blockIdx.x * 256 + threadIdx.x;
    if (i >= HID) return;
    const float ig = 1.0f / (1.0f + expf(-gates[i]));
    const float fg = 1.0f / (1.0f + expf(-gates[i + HID]));
    const float gg = tanhf(gates[i + 2 * HID]);
    const float og = 1.0f / (1.0f + expf(-gates[i + 3 * HID]));
    const float cn = fg * c_old[i] + ig * gg;
    const float hn = og * tanhf(cn);
    h_new[i] = hn; c_new[i] = cn; concat_h[i] = hn; out_h[i] = hn; out_c[i] = cn;
}

// ---------------------------------------------------------------------------
// Single-block softmax over N elements, in place
// ---------------------------------------------------------------------------
__global__ __launch_bounds__(1024) void softmax_inplace(float* __restrict__ s, int N)
{
    __shared__ float red[1024];
    const int tid = threadIdx.x;

    float m = -INFINITY;
    for (int i = tid; i < N; i += 1024) m = fmaxf(m, s[i]);
    red[tid] = m; __syncthreads();
    for (int off = 512; off > 0; off >>= 1) {
        if (tid < off) red[tid] = fmaxf(red[tid], red[tid + off]);
        __syncthreads();
    }
    m = red[0]; __syncthreads();

    float sum = 0.0f;
    for (int i = tid; i < N; i += 1024) sum += expf(s[i] - m);
    red[tid] = sum; __syncthreads();
    for (int off = 512; off > 0; off >>= 1) {
        if (tid < off) red[tid] += red[tid + off];
        __syncthreads();
    }
    const float inv = 1.0f / red[0]; __syncthreads();

    for (int i = tid; i < N; i += 1024) s[i] = expf(s[i] - m) * inv;
}

// ---------------------------------------------------------------------------
// ctx[e] = sum_t attn[t, e] * w[t]  (column access: coalesced VALU FMA; L2-hot)
// ---------------------------------------------------------------------------
__global__ __launch_bounds__(256) void ctx_kernel(const float* __restrict__ attn,
                                                  const float* __restrict__ w,
                                                  float* __restrict__ ctx)
{
    __shared__ float wlds[256];
    const int e = blockIdx.x * 256 + threadIdx.x;   // 0..1023
    float sacc = 0.0f;
    for (int tt = 0; tt < TLEN; tt += 256) {
        __syncthreads();
        wlds[threadIdx.x] = w[tt + threadIdx.x];
        __syncthreads();
#pragma unroll 8
        for (int j = 0; j < 256; ++j)
            sacc = fmaf(attn[(size_t)(tt + j) * HID + e], wlds[j], sacc);
    }
    ctx[e] = sacc;
}

// ---------------------------------------------------------------------------
// Single-block log-softmax: logp[i] = s[i] - max - log(sum(exp(s - max)))
// ---------------------------------------------------------------------------
__global__ __launch_bounds__(1024) void log_softmax_out(const float* __restrict__ s,
                                                        float* __restrict__ outp, int N)
{
    __shared__ float red[1024];
    const int tid = threadIdx.x;

    float m = -INFINITY;
    for (int i = tid; i < N; i += 1024) m = fmaxf(m, s[i]);
    red[tid] = m; __syncthreads();
    for (int off = 512; off > 0; off >>= 1) {
        if (tid < off) red[tid] = fmaxf(red[tid], red[tid + off]);
        __syncthreads();
    }
    m = red[0]; __syncthreads();

    float sum = 0.0f;
    for (int i = tid; i < N; i += 1024) sum += expf(s[i] - m);
    red[tid] = sum; __syncthreads();
    for (int off = 512; off > 0; off >>= 1) {
        if (tid < off) red[tid] += red[tid + off];
        __syncthreads();
    }
    const float shift = m + logf(red[0]);

    for (int i = tid; i < N; i += 1024) outp[i] = s[i] - shift;
}

// ---------------------------------------------------------------------------
extern "C" void kernel_launch(void* const* d_in, const int* in_sizes, int n_in,
                              void* d_out, int out_size, void* d_ws, size_t ws_size,
                              hipStream_t stream)
{
    const float* attn = (const float*)d_in[0];   // (4096, 1, 1024)
    const float* x    = (const float*)d_in[1];   // (1024,)
    const float* h0   = (const float*)d_in[2];   // (1,1,1024)
    const float* c0   = (const float*)d_in[3];   // (1,1,1024)
    const float* W_ih = (const float*)d_in[4];   // (4096, 1024)
    const float* W_hh = (const float*)d_in[5];   // (4096, 1024)
    const float* b_ih = (const float*)d_in[6];   // (4096,)
    const float* b_hh = (const float*)d_in[7];   // (4096,)
    const float* fc1w = (const float*)d_in[8];   // (1024, 2048)
    const float* fc1b = (const float*)d_in[9];   // (1024,)
    const float* fc2w = (const float*)d_in[10];  // (50257, 1024)
    const float* fc2b = (const float*)d_in[11];  // (50257,)
    float* out = (float*)d_out;                  // [logp(50257) | h_new(1024) | c_new(1024)]

    float* ws     = (float*)d_ws;
    float* gates  = ws;            // 4096
    float* h_new  = ws + 4096;     // 1024
    float* c_new  = ws + 5120;     // 1024
    float* probs  = ws + 6144;     // 4096 (scores -> softmax in place)
    float* concat = ws + 10240;    // 2048 = [ctx | h_new]
    float* y1     = ws + 12288;    // 1024
    float* logits = ws + 13312;    // 50257

    const int GB = 128;  // rows per block (8 waves x 16 rows)

    // gates = W_ih@x + b_ih, then += W_hh@h + b_hh
    gemv_wmma<<<(GATES + GB - 1) / GB, 256, 0, stream>>>(W_ih, x,  b_ih, gates, GATES, 1024, 1024, 0);
    gemv_wmma<<<(GATES + GB - 1) / GB, 256, 0, stream>>>(W_hh, h0, b_hh, gates, GATES, 1024, 1024, 1);

    // h_new / c_new (+ write into d_out tail and concat tail)
    lstm_pointwise<<<(HID + 255) / 256, 256, 0, stream>>>(gates, c0, h_new, c_new,
                                                          concat + HID, out + VOCAB, out + VOCAB + HID);

    // scores = attn @ h_new ; softmax ; ctx = attn^T @ w
    gemv_wmma<<<(TLEN + GB - 1) / GB, 256, 0, stream>>>(attn, h_new, nullptr, probs, TLEN, 1024, 1024, 0);
    softmax_inplace<<<1, 1024, 0, stream>>>(probs, TLEN);
    ctx_kernel<<<HID / 256, 256, 0, stream>>>(attn, probs, concat);

    // y1 = fc1_w @ concat + fc1_b ; logits = fc2_w @ y1 + fc2_b
    gemv_wmma<<<(HID + GB - 1) / GB, 256, 0, stream>>>(fc1w, concat, fc1b, y1, HID, 2048, 2048, 0);
    gemv_wmma<<<(VOCAB + GB - 1) / GB, 256, 0, stream>>>(fc2w, y1, fc2b, logits, VOCAB, 1024, 1024, 0);

    // logp into d_out head
    log_softmax_out<<<1, 1024, 0, stream>>>(logits, out, VOCAB);
}